// GRU_ODEINT_EULER_16707422781841
// MI455X (gfx1250) — compile-verified
//
#include <hip/hip_runtime.h>

typedef float v2f __attribute__((ext_vector_type(2)));
typedef float v8f __attribute__((ext_vector_type(8)));

#define NB 2048
#define NF 32
#define NH 64
#define NT 1000
#define G3 192          // 3*H
#define XS 34           // padded LDS row stride for x (floats)
#define HS 66           // padded LDS row stride for h
#define WIS 34          // W_ih row stride
#define WHS 66          // W_hh row stride
#define WLS 66          // W_lin row stride

static __device__ __forceinline__ v8f splat8(float v) {
  v8f r;
#pragma unroll
  for (int i = 0; i < 8; ++i) r[i] = v;
  return r;
}

static __device__ __forceinline__ v8f wmma_f32(v2f a, v2f b, v8f c) {
  // D = A(16x4 f32) x B(4x16 f32) + C(16x16 f32), wave32
  return __builtin_amdgcn_wmma_f32_16x16x4_f32(false, a, false, b, (short)0, c,
                                               false, false);
}

__launch_bounds__(128, 1)
__global__ void gru_ode_euler_kernel(const float* __restrict__ x_in,
                                     const float* __restrict__ tvec,
                                     const float* __restrict__ W_ih,
                                     const float* __restrict__ W_hh,
                                     const float* __restrict__ b_ih,
                                     const float* __restrict__ b_hh,
                                     const float* __restrict__ W_lin,
                                     const float* __restrict__ b_lin,
                                     float* __restrict__ out) {
  __shared__ __attribute__((aligned(16))) float sWih[G3 * WIS];
  __shared__ __attribute__((aligned(16))) float sWhh[G3 * WHS];
  __shared__ __attribute__((aligned(16))) float sWlin[NF * WLS];
  __shared__ float sbI[G3];
  __shared__ float sbH[G3];
  __shared__ float sbL[NF];
  __shared__ __attribute__((aligned(16))) float sX[2][16 * XS];
  __shared__ __attribute__((aligned(16))) float sH[2][16 * HS];

  const int tid  = threadIdx.x;     // 0..127
  const int lane = tid & 31;
  const int w    = tid >> 5;        // wave id == 16-wide hidden/gate slice
  const int l15  = lane & 15;
  const int hf   = lane >> 4;       // half-wave selector (K pair / M+8 rows)
  const int base = blockIdx.x * 16; // batch-row tile base

  // ---- one-time staging: weights + biases into LDS (padded strides) ----
  for (int i = tid; i < G3 * NF; i += 128)
    sWih[(i >> 5) * WIS + (i & 31)] = W_ih[i];
  for (int i = tid; i < G3 * NH; i += 128)
    sWhh[(i >> 6) * WHS + (i & 63)] = W_hh[i];
  for (int i = tid; i < NF * NH; i += 128)
    sWlin[(i >> 6) * WLS + (i & 63)] = W_lin[i];
  for (int i = tid; i < G3; i += 128) { sbI[i] = b_ih[i]; sbH[i] = b_hh[i]; }
  if (tid < NF) sbL[tid] = b_lin[tid];

  // x0 tile -> LDS buffer 0 and out[t=0]; h0 = 0
  for (int i = tid; i < 16 * NF; i += 128) {
    float v = x_in[(base + (i >> 5)) * NF + (i & 31)];
    sX[0][(i >> 5) * XS + (i & 31)] = v;
    out[(size_t)(base + (i >> 5)) * NF + (i & 31)] = v;
  }
  for (int i = tid; i < 16 * XS; i += 128) sX[1][i] = 0.f;
  for (int i = tid; i < 16 * HS; i += 128) { sH[0][i] = 0.f; sH[1][i] = 0.f; }
  __syncthreads();

  const int off = w * 16 + l15; // gate column within a 64-wide gate block

  for (int s = 0; s < NT - 1; ++s) {
    const int cur = s & 1, nxt = cur ^ 1;
    const float dt = (s == 0) ? (tvec[1] - tvec[0]) : (tvec[s] - tvec[s - 1]);

    // A-matrix fragments: lane l15 = row M, half-wave picks K pair (b64 loads)
    v2f xa[8];
#pragma unroll
    for (int k = 0; k < 8; ++k)
      xa[k] = *(const v2f*)&sX[cur][l15 * XS + 4 * k + 2 * hf];
    v2f ha[16];
#pragma unroll
    for (int k = 0; k < 16; ++k)
      ha[k] = *(const v2f*)&sH[cur][l15 * HS + 4 * k + 2 * hf];

    // C init = biases broadcast along M (bias depends only on N = lane&15)
    v8f accR  = splat8(sbI[off] + sbH[off]);                 // r gate: rows 0..63
    v8f accZ  = splat8(sbI[NH + off] + sbH[NH + off]);       // z gate: rows 64..127
    v8f accNi = splat8(sbI[2 * NH + off]);                   // n gate, input part
    v8f accNh = splat8(sbH[2 * NH + off]);                   // n gate, recurrent part

    // gi contributions: x(16x32) x W_ih^T slices, K in chunks of 4
#pragma unroll
    for (int k = 0; k < 8; ++k) {
      const int c = 4 * k + 2 * hf;
      v2f bR = *(const v2f*)&sWih[(off) * WIS + c];
      v2f bZ = *(const v2f*)&sWih[(64 + off) * WIS + c];
      v2f bN = *(const v2f*)&sWih[(128 + off) * WIS + c];
      accR  = wmma_f32(xa[k], bR, accR);
      accZ  = wmma_f32(xa[k], bZ, accZ);
      accNi = wmma_f32(xa[k], bN, accNi);
    }
    // gh contributions: h(16x64) x W_hh^T slices
#pragma unroll
    for (int k = 0; k < 16; ++k) {
      const int c = 4 * k + 2 * hf;
      v2f bR = *(const v2f*)&sWhh[(off) * WHS + c];
      v2f bZ = *(const v2f*)&sWhh[(64 + off) * WHS + c];
      v2f bN = *(const v2f*)&sWhh[(128 + off) * WHS + c];
      accR  = wmma_f32(ha[k], bR, accR);
      accZ  = wmma_f32(ha[k], bZ, accZ);
      accNh = wmma_f32(ha[k], bN, accNh);
    }

    // gate nonlinearity in C/D layout: VGPR j -> row (j + 8*hf), col = off
#pragma unroll
    for (int j = 0; j < 8; ++j) {
      const int row = j + 8 * hf;
      const float ho = sH[cur][row * HS + off];
      const float r  = 1.f / (1.f + __expf(-accR[j]));
      const float z  = 1.f / (1.f + __expf(-accZ[j]));
      const float n  = tanhf(accNi[j] + r * accNh[j]);
      sH[nxt][row * HS + off] = (1.f - z) * n + z * ho;
    }
    __syncthreads(); // h_new fully visible before dx reads it

    // dx = h_new x W_lin^T (16x32): waves 0,1 own the two 16-wide f slices
    if (w < 2) {
      v8f accX = splat8(sbL[off]);
#pragma unroll
      for (int k = 0; k < 16; ++k) {
        const int c = 4 * k + 2 * hf;
        v2f a = *(const v2f*)&sH[nxt][l15 * HS + c];
        v2f b = *(const v2f*)&sWlin[off * WLS + c];
        accX = wmma_f32(a, b, accX);
      }
      float* orow = out + (size_t)(s + 1) * (NB * NF);
#pragma unroll
      for (int j = 0; j < 8; ++j) {
        const int row = j + 8 * hf;
        const float xn = sX[cur][row * XS + off] + accX[j] * dt;
        sX[nxt][row * XS + off] = xn;
        orow[(size_t)(base + row) * NF + off] = xn; // lanes 0-15 contiguous in f
      }
    }
    __syncthreads(); // x_new visible for next step's A-frag loads
  }
}

extern "C" void kernel_launch(void* const* d_in, const int* in_sizes, int n_in,
                              void* d_out, int out_size, void* d_ws, size_t ws_size,
                              hipStream_t stream) {
  const float* x_in  = (const float*)d_in[0];
  const float* t     = (const float*)d_in[1];
  const float* W_ih  = (const float*)d_in[2];
  const float* W_hh  = (const float*)d_in[3];
  const float* b_ih  = (const float*)d_in[4];
  const float* b_hh  = (const float*)d_in[5];
  const float* W_lin = (const float*)d_in[6];
  const float* b_lin = (const float*)d_in[7];
  float* out = (float*)d_out;

  dim3 grid(NB / 16);   // 128 workgroups, one 16-row batch tile each
  dim3 block(128);      // 4 wave32 -> 4 hidden slices
  hipLaunchKernelGGL(gru_ode_euler_kernel, grid, block, 0, stream,
                     x_in, t, W_ih, W_hh, b_ih, b_hh, W_lin, b_lin, out);
}